// ARAPLoss_4776003633585
// MI455X (gfx1250) — compile-verified
//
#include <hip/hip_runtime.h>

// ---------------------------------------------------------------------------
// ARAP-style edge loss:  mean_e | ||x[d]-x[s]||^2 - ||dx[d]-dx[s]||^2 |
//
// MI455X strategy:
//  * x/dx (48MB) stay resident in the 192MB L2; the random 12B gathers are
//    L2 hits -> optimize for memory-level parallelism (16 independent gathers
//    in flight per thread per chunk).
//  * The 192MB read-once edge-index streams go over the CDNA5 async-to-LDS
//    path (global_load_async_to_lds_b128, ASYNCcnt) with an NT temporal hint:
//    decoupled from the gather LOADcnt stream and no L2 pollution.
//  * Each lane async-copies exactly the 16B of indices it consumes itself,
//    so no __syncthreads is needed; only per-wave s_wait_asynccnt.
//  * Deterministic reduction: wave32 butterfly -> LDS -> per-block partial ->
//    fixed-order single-block finalize (no float atomics; graph-replay safe).
// ---------------------------------------------------------------------------

typedef int v4i __attribute__((ext_vector_type(4)));

struct Float3 { float x, y, z; };

__device__ __forceinline__ Float3 load3(const float* __restrict__ p) {
    Float3 r;
    r.x = p[0];
    r.y = p[1];
    r.z = p[2];
    return r;
}

__device__ __forceinline__ float edge_term(const float* __restrict__ x,
                                           const float* __restrict__ dx,
                                           int s, int d) {
    const Float3 xs = load3(x  + 3 * s);
    const Float3 xd = load3(x  + 3 * d);
    const Float3 vs = load3(dx + 3 * s);
    const Float3 vd = load3(dx + 3 * d);
    const float ax = xd.x - xs.x, ay = xd.y - xs.y, az = xd.z - xs.z;
    const float bx = vd.x - vs.x, by = vd.y - vs.y, bz = vd.z - vs.z;
    const float diffx  = fmaf(ax, ax, fmaf(ay, ay, az * az));
    const float diffdx = fmaf(bx, bx, fmaf(by, by, bz * bz));
    return fabsf(diffx - diffdx);
}

#define BLOCK_THREADS 256
#define EDGES_PER_THREAD 4
#define CHUNK_EDGES (BLOCK_THREADS * EDGES_PER_THREAD)   // 1024 edges / chunk

__global__ void __launch_bounds__(BLOCK_THREADS)
arap_partial_kernel(const float* __restrict__ dx,
                    const float* __restrict__ x,
                    const int*   __restrict__ esrc,
                    const int*   __restrict__ edst,
                    float*       __restrict__ partial,
                    int E) {
    // Double-buffered index staging: 2 arrays * 2 buffers * 4KB = 16KB LDS.
    __shared__ v4i s_src[2][BLOCK_THREADS];
    __shared__ v4i s_dst[2][BLOCK_THREADS];

    const int tid = threadIdx.x;
    float acc = 0.0f;

    const int nchunks = E / CHUNK_EDGES;
    const int cstride = gridDim.x;

    // Issue one chunk's index copy on the CDNA5 async-to-LDS path (NT hint:
    // read-once stream must not evict the L2-resident x/dx working set).
    auto issue_chunk = [&](int chunk, int buf) {
        const unsigned byteoff =
            ((unsigned)chunk * (unsigned)CHUNK_EDGES + (unsigned)tid * 4u) * 4u;
        const unsigned lds_s = (unsigned)(uintptr_t)&s_src[buf][tid];
        const unsigned lds_d = (unsigned)(uintptr_t)&s_dst[buf][tid];
        asm volatile("global_load_async_to_lds_b128 %0, %1, %2 th:TH_LOAD_NT"
                     :: "v"(lds_s), "v"(byteoff), "s"(esrc) : "memory");
        asm volatile("global_load_async_to_lds_b128 %0, %1, %2 th:TH_LOAD_NT"
                     :: "v"(lds_d), "v"(byteoff), "s"(edst) : "memory");
    };

    int c = blockIdx.x;
    if (c < nchunks) {
        int buf = 0;
        issue_chunk(c, buf);                       // prologue: chunk 0 in flight
        for (; c < nchunks; c += cstride) {
            const int  cn   = c + cstride;
            const bool more = cn < nchunks;
            if (more) {
                issue_chunk(cn, buf ^ 1);          // keep next chunk in flight
                // 2 older + 2 newer outstanding; async ops complete in order,
                // so <=2 means the current chunk's two copies have landed.
                asm volatile("s_wait_asynccnt 0x2" ::: "memory");
            } else {
                asm volatile("s_wait_asynccnt 0x0" ::: "memory");
            }
            // Each lane reads back exactly the 16B it copied -> no barrier.
            const v4i s4 = s_src[buf][tid];
            const v4i d4 = s_dst[buf][tid];
            acc += edge_term(x, dx, s4.x, d4.x);
            acc += edge_term(x, dx, s4.y, d4.y);
            acc += edge_term(x, dx, s4.z, d4.z);
            acc += edge_term(x, dx, s4.w, d4.w);
            buf ^= 1;
        }
    }

    // Tail (< CHUNK_EDGES edges): plain gathers, grid-strided.
    const int tail_start = nchunks * CHUNK_EDGES;
    const int gtid       = blockIdx.x * BLOCK_THREADS + tid;
    for (int t = tail_start + gtid; t < E; t += gridDim.x * BLOCK_THREADS)
        acc += edge_term(x, dx, esrc[t], edst[t]);

    // Wave32 butterfly reduction (CDNA5 is wave32: 5 steps).
    #pragma unroll
    for (int off = 16; off > 0; off >>= 1)
        acc += __shfl_xor(acc, off, 32);

    __shared__ float smem[BLOCK_THREADS / 32];
    const int lane = tid & 31;
    const int wave = tid >> 5;
    if (lane == 0) smem[wave] = acc;
    __syncthreads();

    if (tid == 0) {
        float s = 0.0f;
        #pragma unroll
        for (int w = 0; w < BLOCK_THREADS / 32; ++w)
            s += smem[w];
        partial[blockIdx.x] = s;                   // deterministic, no atomics
    }
}

__global__ void __launch_bounds__(256)
arap_finalize_kernel(const float* __restrict__ partial,
                     int nblk,
                     float* __restrict__ out,
                     float invE) {
    __shared__ float smem[256];
    const int tid = threadIdx.x;
    float s = 0.0f;
    for (int i = tid; i < nblk; i += 256)          // fixed traversal order
        s += partial[i];
    smem[tid] = s;
    __syncthreads();
    #pragma unroll
    for (int stride = 128; stride > 0; stride >>= 1) {
        if (tid < stride) smem[tid] += smem[tid + stride];
        __syncthreads();
    }
    if (tid == 0) out[0] = smem[0] * invE;
}

extern "C" void kernel_launch(void* const* d_in, const int* in_sizes, int n_in,
                              void* d_out, int out_size, void* d_ws, size_t ws_size,
                              hipStream_t stream) {
    // setup_inputs() order: dx, x, edge_src, edge_dst
    const float* dx   = (const float*)d_in[0];
    const float* x    = (const float*)d_in[1];
    const int*   esrc = (const int*)d_in[2];
    const int*   edst = (const int*)d_in[3];
    const int    E    = in_sizes[2];

    float* partial = (float*)d_ws;

    int nblk = 4096;
    const int maxblk = (int)(ws_size / sizeof(float));
    if (maxblk > 0 && nblk > maxblk) nblk = maxblk;
    const int nchunks = E / CHUNK_EDGES;
    if (nchunks > 0 && nblk > nchunks) nblk = nchunks;
    if (nblk < 1) nblk = 1;

    arap_partial_kernel<<<nblk, BLOCK_THREADS, 0, stream>>>(dx, x, esrc, edst,
                                                            partial, E);

    const float invE = (E > 0) ? (1.0f / (float)E) : 0.0f;
    arap_finalize_kernel<<<1, 256, 0, stream>>>(partial, nblk, (float*)d_out, invE);
}